// CustomCNNLayer_84559316124470
// MI455X (gfx1250) — compile-verified
//
#include <hip/hip_runtime.h>

typedef float v2f __attribute__((ext_vector_type(2)));
typedef float v8f __attribute__((ext_vector_type(8)));
typedef int   v4i __attribute__((ext_vector_type(4)));

#define S_LEN     4096
#define C_IN      2
#define K_WIN     64
#define NK_       32
#define OUT_CH    512
#define KDIM      4096   // contraction length: C_IN * 2048
#define LDS_XP    4176   // padded per-channel pitch: max p = 2*2047+79 = 4173
#define KSTEPS    256    // per-wave WMMA iterations: (KDIM/4 waves)/4 K-per-wmma

#if defined(__gfx1250__) && \
    __has_builtin(__builtin_amdgcn_global_load_async_to_lds_b128) && \
    __has_builtin(__builtin_amdgcn_s_wait_asynccnt)
#define ASYNC_COPY 1
typedef __attribute__((address_space(1))) v4i gv4i;   // prints as "__device__"
typedef __attribute__((address_space(3))) v4i lv4i;   // prints as "__shared__"
#else
#define ASYNC_COPY 0
#endif

// out[b,o,t] = relu(conv_b[o] + stft_w[l,m] * G[b,o,m+r]),  t = (r,m,l)
// G[b,o,u]   = sum_{c,q} conv_w[o,c*2048+q] * xp[b,c,2q+u]  (WMMA f32 GEMM)
// Block = 128 threads (4 waves) owns one 16(o) x 16(u) tile of G.
// x[b] staged in LDS (zero-padded -> branch-free B fetch); each wave takes
// K/4 of the contraction (half a channel each); LDS reduction; direct expand.
__global__ __launch_bounds__(128)
void fused_stft_conv_kernel(const float* __restrict__ x,
                            const float* __restrict__ stft_w,
                            const float* __restrict__ conv_w,
                            const float* __restrict__ conv_b,
                            float* __restrict__ out) {
  __shared__ __align__(16) float lds_x[C_IN * LDS_XP];  // 33,408 B
  __shared__ float lds_acc[4 * 256];                    // wave partial tiles
  __shared__ float lds_w[K_WIN * NK_];                  // stft_w transposed

  const int ut   = blockIdx.x;            // u tile: 0..4
  const int mt   = blockIdx.y;            // o tile: 0..31
  const int b    = blockIdx.z;            // batch: 0..3
  const int tid  = threadIdx.x;
  const int lane = tid & 31;
  const int wv   = tid >> 5;              // wave id 0..3

  const int m  = lane & 15;               // A row / B,C col within tile
  const int hf = lane >> 4;               // lane half: K sub-pair select
  const int o0 = mt * 16;
  const int u0 = ut * 16;

  const float* xb = x + (size_t)b * (C_IN * S_LEN);

  const int k_begin = wv * (KDIM / 4);            // wave K range start
  const float* aptr = conv_w + (size_t)(o0 + m) * KDIM + k_begin + hf * 2;
  __builtin_prefetch(aptr, 0, 1);                 // global_prefetch_b8 warm-up

  // ---- Stage x[b] into LDS (async global->LDS on CDNA5) ----
  for (int g4 = tid; g4 < (C_IN * S_LEN) / 4; g4 += 128) {
    const int c  = g4 >> 10;                      // channel
    const int p4 = g4 & 1023;                     // float4 index in channel
    float* ldst = &lds_x[c * LDS_XP + (p4 << 2)];
#if ASYNC_COPY
    __builtin_amdgcn_global_load_async_to_lds_b128(
        (gv4i*)(xb + (g4 << 2)), (lv4i*)ldst, 0, 0);
#else
    *(float4*)ldst = ((const float4*)xb)[g4];
#endif
  }
  // Zero the pad tails (the jnp.pad region): p in [4096, 4176) per channel.
  for (int idx = tid; idx < C_IN * (LDS_XP - S_LEN); idx += 128) {
    const int c   = idx >= (LDS_XP - S_LEN);
    const int off = idx - c * (LDS_XP - S_LEN);
    lds_x[c * LDS_XP + S_LEN + off] = 0.f;
  }
  // Transposed stft_w: lds_w[m][l] for conflict-free lane reads in expand.
  for (int idx = tid; idx < K_WIN * NK_; idx += 128) {
    lds_w[(idx & 63) * NK_ + (idx >> 6)] = stft_w[idx];
  }
#if ASYNC_COPY
  __builtin_amdgcn_s_wait_asynccnt(0);
#endif
  __syncthreads();

  // ---- WMMA main loop: branch-free, A pipelined one iteration ahead ----
  // B fetch: i0 = k + hf*2 (even), i1 = i0+1 always same channel;
  // lds index = c*LDS_XP + 2*(i0 - c*2048) + u0 + m, advancing +8 per step.
  const int cw = k_begin >> 11;                   // this wave's channel
  int ldsi = cw * LDS_XP + 2 * (k_begin + hf * 2 - cw * 2048) + u0 + m;

  v8f acc = {0.f, 0.f, 0.f, 0.f, 0.f, 0.f, 0.f, 0.f};
  v2f av; av.x = aptr[0]; av.y = aptr[1];

  #pragma unroll 4
  for (int it = 0; it < KSTEPS - 1; ++it) {
    v2f bv; bv.x = lds_x[ldsi]; bv.y = lds_x[ldsi + 2];   // ds_load_2addr
    aptr += 4;
    v2f avn; avn.x = aptr[0]; avn.y = aptr[1];            // next A in flight
    acc = __builtin_amdgcn_wmma_f32_16x16x4_f32(
        false, av, false, bv, (short)0, acc, false, false);
    av = avn;
    ldsi += 8;
  }
  { // epilogue (no A over-read past the wave's K range)
    v2f bv; bv.x = lds_x[ldsi]; bv.y = lds_x[ldsi + 2];
    acc = __builtin_amdgcn_wmma_f32_16x16x4_f32(
        false, av, false, bv, (short)0, acc, false, false);
  }

  // C/D layout: vgpr j -> row (j + hf*8), col m.
  #pragma unroll
  for (int j = 0; j < 8; ++j)
    lds_acc[wv * 256 + (j + hf * 8) * 16 + m] = acc[j];

  __syncthreads();

  // ---- Reduce 4 wave partials, expand directly to out (BW-bound) ----
  const int l  = tid & 31;                // streams the NK dimension
  const int gg = tid >> 5;

  for (int e = gg; e < 256; e += 4) {
    const int row = e >> 4;
    const int col = e & 15;
    const int o   = o0 + row;
    const int u   = u0 + col;
    const float g = lds_acc[e] + lds_acc[256 + e] +
                    lds_acc[512 + e] + lds_acc[768 + e];
    const float bias = conv_b[o];
    #pragma unroll
    for (int r = 0; r < 2; ++r) {
      const int mm = u - r;               // valid iff 0 <= mm < 64
      if ((unsigned)mm < (unsigned)K_WIN) {
        const float w = lds_w[mm * NK_ + l];
        float v = bias + w * g;
        v = v > 0.f ? v : 0.f;
        const size_t t = (size_t)(r * 2048 + mm * 32 + l);
        __builtin_nontemporal_store(
            v, out + (((size_t)b * OUT_CH + o) << 12) + t);
      }
    }
  }
}

extern "C" void kernel_launch(void* const* d_in, const int* in_sizes, int n_in,
                              void* d_out, int out_size, void* d_ws, size_t ws_size,
                              hipStream_t stream) {
  (void)in_sizes; (void)n_in; (void)d_ws; (void)ws_size; (void)out_size;
  const float* x      = (const float*)d_in[0];   // (4, 2, 4096)
  const float* stft_w = (const float*)d_in[1];   // (32, 64)
  const float* conv_w = (const float*)d_in[2];   // (512, 4096)
  const float* conv_b = (const float*)d_in[3];   // (512,)
  float* out = (float*)d_out;                    // (4, 512, 4096)

  dim3 grid(5, 32, 4);                           // u-tiles x o-tiles x batch
  fused_stft_conv_kernel<<<grid, 128, 0, stream>>>(x, stft_w, conv_w, conv_b, out);
}